// ParsingRelationDis_69234872811936
// MI455X (gfx1250) — compile-verified
//
#include <hip/hip_runtime.h>
#include <hip/hip_bf16.h>
#include <stdint.h>

#define N_DIM   101
#define G_CLS   100
#define R_LEN   56
#define L_LANES 4
#define P_POS   224   // R_LEN * L_LANES
#define CHUNK   32    // channel rows per async chunk
#define TAIL    4     // 100 - 3*32

// ---------------- CDNA5 async global->LDS helpers ----------------

__device__ __forceinline__ unsigned lds_off(const void* p) {
  // Low 32 bits of a flat pointer into LDS == LDS byte offset (LDS aperture).
  return (unsigned)(uintptr_t)p;
}

#define AS1 __attribute__((address_space(1)))
#define AS3 __attribute__((address_space(3)))

typedef int v4i __attribute__((ext_vector_type(4)));
typedef AS1 v4i* gv4i_p;   // global address space src
typedef AS3 v4i* lv4i_p;   // LDS address space dst

__device__ __forceinline__ void async_b128(const void* g, unsigned lds_byte_off) {
#if __has_builtin(__builtin_amdgcn_global_load_async_to_lds_b128)
  __builtin_amdgcn_global_load_async_to_lds_b128(
      (gv4i_p)(uintptr_t)g,
      (lv4i_p)lds_byte_off,
      0, 0);
#else
  unsigned long long ga = (unsigned long long)(uintptr_t)g;
  asm volatile("global_load_async_to_lds_b128 %0, %1, off"
               :: "v"(lds_byte_off), "v"(ga) : "memory");
#endif
}

#if __has_builtin(__builtin_amdgcn_s_wait_asynccnt)
#define WAIT_ASYNC(n) __builtin_amdgcn_s_wait_asynccnt(n)
#else
#define WAIT_ASYNC(n) asm volatile("s_wait_asynccnt " #n ::: "memory")
#endif

// Issue one full 32-row chunk: 32*224 floats = 1792 x b128, 7 per thread.
__device__ __forceinline__ void issue_full(const float* gbase, unsigned ldsbase, int t) {
#pragma unroll
  for (int k = 0; k < 7; ++k) {
    const int f = t + k * 256;                 // float4 index
    async_b128(gbase + (size_t)f * 4, ldsbase + (unsigned)f * 16u);
  }
}

// Tail chunk: 4*224 floats = 224 x b128. Pad wave 7 with a dummy transfer so
// every wave's ASYNCcnt increments by exactly 1 (uniform s_wait immediates).
__device__ __forceinline__ void issue_tail(const float* gbase, unsigned ldsbase,
                                           unsigned dummyOff, int t) {
  const bool act = (t < P_POS);
  const int f = act ? t : 0;
  const unsigned l = act ? (ldsbase + (unsigned)t * 16u) : dummyOff;
  async_b128(gbase + (size_t)f * 4, l);
}

template <int ROWS>
__device__ __forceinline__ void accum_rows(const float* bp, int t, int gbase,
                                           float& s, float& w) {
#pragma unroll
  for (int r = 0; r < ROWS; ++r) {
    const float v = bp[r * P_POS + t];   // ds_load_b32, conflict-free
    const float e = __expf(v);           // v_exp_f32
    s += e;
    w = fmaf((float)(gbase + r), e, w);
  }
}

// ---------------- main kernel: one block per sample ----------------

__global__ __launch_bounds__(256) void
parse_loss_kernel(const float* __restrict__ x, const int* __restrict__ labels,
                  float* __restrict__ per_sample) {
  __shared__ alignas(16) float buf[2][CHUNK * P_POS];  // 2 x 28672 B
  __shared__ alignas(16) float dummyPad[4];
  __shared__ float posS[P_POS];
  __shared__ unsigned char validS[P_POS];
  __shared__ float laneLoss[L_LANES];

  const int n = blockIdx.x;
  const int t = threadIdx.x;
  const float* xb = x + (size_t)n * N_DIM * P_POS;

  if (t < P_POS) {
    validS[t] = (labels[(size_t)n * P_POS + t] < G_CLS) ? 1 : 0;
  }

  const unsigned b0  = lds_off(&buf[0][0]);
  const unsigned b1  = lds_off(&buf[1][0]);
  const unsigned dOf = lds_off(&dummyPad[0]);

  float s = 0.f, w = 0.f;

  // Double-buffered async pipeline over channel chunks [32,32,32,4].
  issue_full(xb, b0, t);                                     // chunk0 -> buf0

  issue_full(xb + (size_t)1 * CHUNK * P_POS, b1, t);         // chunk1 -> buf1
  WAIT_ASYNC(7);                                             // chunk0 done
  __syncthreads();
  if (t < P_POS) accum_rows<CHUNK>(&buf[0][0], t, 0, s, w);
  __syncthreads();

  issue_full(xb + (size_t)2 * CHUNK * P_POS, b0, t);         // chunk2 -> buf0
  WAIT_ASYNC(7);                                             // chunk1 done
  __syncthreads();
  if (t < P_POS) accum_rows<CHUNK>(&buf[1][0], t, 32, s, w);
  __syncthreads();

  issue_tail(xb + (size_t)3 * CHUNK * P_POS, b1, dOf, t);    // chunk3 -> buf1
  WAIT_ASYNC(1);                                             // chunk2 done
  __syncthreads();
  if (t < P_POS) accum_rows<CHUNK>(&buf[0][0], t, 64, s, w);
  __syncthreads();

  WAIT_ASYNC(0);                                             // chunk3 done
  __syncthreads();
  if (t < P_POS) {
    accum_rows<TAIL>(&buf[1][0], t, 96, s, w);
    posS[t] = w / s;                                         // softmax expectation
  }
  __syncthreads();

  // Per-lane bounds + masked second-difference loss (4 threads).
  if (t < L_LANES) {
    const int l = t;
    const bool v0    = validS[0 * L_LANES + l] != 0;
    const bool v1    = validS[1 * L_LANES + l] != 0;
    const bool vlast = validS[(R_LEN - 1) * L_LANES + l] != 0;
    bool has_up = false, has_dn = false;
    int first_up = 0, first_dn = 0;
    for (int r = 0; r < R_LEN - 1; ++r) {
      const bool a = validS[r * L_LANES + l] != 0;
      const bool b = validS[(r + 1) * L_LANES + l] != 0;
      if (!a && b && !has_up) { has_up = true; first_up = r + 1; }
      if (a && !b && !has_dn) { has_dn = true; first_dn = r; }
    }
    int top = v0 ? 0 : (has_up ? first_up : 0);
    const bool top_exists = v0 || has_up;
    const bool cond0 = v0 && !v1;
    int down = cond0 ? 0 : (vlast ? (R_LEN - 1) : (has_dn ? first_dn : 0));
    const bool down_exists = cond0 || vlast || has_dn;
    if (!(top_exists && down_exists)) { top = 0; down = 0; }
    const int count = down - top - 1;

    float sum = 0.f;
    for (int idx = 0; idx < R_LEN - 2; ++idx) {
      if (idx >= top && idx <= down - 2) {
        const float p0 = posS[idx * L_LANES + l];
        const float p1 = posS[(idx + 1) * L_LANES + l];
        const float p2 = posS[(idx + 2) * L_LANES + l];
        sum += fabsf((p0 - p1) - (p1 - p2));   // |d[i] - d[i+1]|
      }
    }
    laneLoss[l] = (count >= 1) ? (sum / (float)count) : 0.f;
  }
  __syncthreads();

  if (t == 0) {
    per_sample[n] =
        (laneLoss[0] + laneLoss[1] + laneLoss[2] + laneLoss[3]) * 0.125f;
  }
}

// ---------------- deterministic final reduction ----------------

__global__ __launch_bounds__(256) void
final_reduce_kernel(const float* __restrict__ per_sample, float* __restrict__ out,
                    int N) {
  __shared__ float st[256];
  __shared__ float sf[256];
  float tsum = 0.f, fsum = 0.f;
  for (int i = threadIdx.x; i < N; i += 256) {
    const float v = per_sample[i];
    tsum += v;
    fsum += (v != 0.f) ? 1.f : 0.f;
  }
  st[threadIdx.x] = tsum;
  sf[threadIdx.x] = fsum;
  __syncthreads();
  for (int s2 = 128; s2 > 0; s2 >>= 1) {
    if (threadIdx.x < (unsigned)s2) {
      st[threadIdx.x] += st[threadIdx.x + s2];
      sf[threadIdx.x] += sf[threadIdx.x + s2];
    }
    __syncthreads();
  }
  if (threadIdx.x == 0) {
    out[0] = (sf[0] > 0.f) ? (st[0] / sf[0]) : 0.f;
  }
}

// ---------------- entry ----------------

extern "C" void kernel_launch(void* const* d_in, const int* in_sizes, int n_in,
                              void* d_out, int out_size, void* d_ws, size_t ws_size,
                              hipStream_t stream) {
  const float* x      = (const float*)d_in[0];
  const int*   labels = (const int*)d_in[1];
  float* out          = (float*)d_out;
  float* per_sample   = (float*)d_ws;   // N floats of scratch

  const int N = in_sizes[1] / (R_LEN * L_LANES);   // 4096

  parse_loss_kernel<<<dim3(N), dim3(256), 0, stream>>>(x, labels, per_sample);
  final_reduce_kernel<<<dim3(1), dim3(256), 0, stream>>>(per_sample, out, N);
}